// GCN_17497696764522
// MI455X (gfx1250) — compile-verified
//
#include <hip/hip_runtime.h>

typedef __attribute__((ext_vector_type(2))) float v2f;
typedef __attribute__((ext_vector_type(8))) float v8f;

#define IN_DIM 256
#define HID    16

// ---------------- degree / norm ----------------

__global__ void k_init_deg(float* __restrict__ deg, int n) {
    int i = blockIdx.x * blockDim.x + threadIdx.x;
    if (i < n) deg[i] = 1.0f;                // self-loop weight
}

__global__ void k_deg(const int* __restrict__ dst, const float* __restrict__ w,
                      float* __restrict__ deg, int e) {
    int i = blockIdx.x * blockDim.x + threadIdx.x;
    if (i < e) atomicAdd(&deg[dst[i]], w[i]);
}

__global__ void k_rsqrt(float* __restrict__ deg, int n) {
    int i = blockIdx.x * blockDim.x + threadIdx.x;
    if (i < n) deg[i] = __frsqrt_rn(deg[i]); // deg >= 1 always (self loop)
}

__global__ void k_norm(const int* __restrict__ src, const int* __restrict__ dst,
                       const float* __restrict__ w, const float* __restrict__ dis,
                       float* __restrict__ norm, int e) {
    int i = blockIdx.x * blockDim.x + threadIdx.x;
    if (i < e) norm[i] = dis[src[i]] * w[i] * dis[dst[i]];
}

// ---------------- layer-1 dense transform: h = x @ W1 (WMMA f32 16x16x4) ----------------
// One wave per 16-row tile of x. W1 staged transposed in LDS: w1t[n][k] (16x256 f32).

__global__ void k_gemm1(const float* __restrict__ x, const float* __restrict__ W1,
                        float* __restrict__ h, int n_tiles) {
    __shared__ float w1t[HID * IN_DIM];      // 16 KB
    for (int i = threadIdx.x; i < HID * IN_DIM; i += blockDim.x) {
        int n = i >> 8;                      // 0..15
        int k = i & 255;                     // 0..255
        w1t[i] = W1[k * HID + n];
    }
    __syncthreads();

    int wave = blockIdx.x * (blockDim.x >> 5) + (threadIdx.x >> 5);
    if (wave >= n_tiles) return;             // wave-uniform: EXEC stays all-ones
    int lane = threadIdx.x & 31;
    int half = lane >> 4;                    // 0: K pair (k,k+1)   1: K pair (k+2,k+3)
    int l    = lane & 15;                    // A: row   B: column

    const float* arow = x + (size_t)(wave * 16 + l) * IN_DIM + half * 2;
    const float* brow = &w1t[l * IN_DIM + half * 2];

    v8f acc = {};
#pragma unroll 8
    for (int c = 0; c < IN_DIM / 4; ++c) {
        int k = c * 4;
        v2f a = *(const v2f*)(arow + k);     // global_load_b64
        v2f b = *(const v2f*)(brow + k);     // ds_load_b64
        acc = __builtin_amdgcn_wmma_f32_16x16x4_f32(
                  false, a, false, b, (short)0, acc, false, false);
    }

    // C/D layout: VGPR r -> lanes 0-15: M=r, N=lane ; lanes 16-31: M=r+8, N=lane-16
    float* outp = h + (size_t)(wave * 16 + half * 8) * HID + l;
#pragma unroll
    for (int r = 0; r < 8; ++r) outp[r * HID] = acc[r];
}

// ---------------- layer-1 aggregation ----------------
// agg = h*dis^2 + b1 (self loop + bias, full overwrite), then edge scatter, then ReLU.

__global__ void k_agg1_init(const float* __restrict__ h, const float* __restrict__ dis,
                            const float* __restrict__ b1, float* __restrict__ agg, int n16) {
    int t = blockIdx.x * blockDim.x + threadIdx.x;
    if (t < n16) {
        int i = t >> 4, j = t & 15;
        float d = dis[i];
        agg[t] = h[t] * d * d + b1[j];
    }
}

__global__ void k_agg1(const int* __restrict__ src, const int* __restrict__ dst,
                       const float* __restrict__ norm, const float* __restrict__ h,
                       float* __restrict__ agg, int e) {
    int t = blockIdx.x * blockDim.x + threadIdx.x;   // e*16 threads, 16 lanes per edge
    if (t < e * 16) {
        int ei = t >> 4, j = t & 15;
        int s = src[ei], d = dst[ei];
        atomicAdd(&agg[d * HID + j], h[s * HID + j] * norm[ei]);
    }
}

__global__ void k_relu(float* __restrict__ a, int n16) {
    int t = blockIdx.x * blockDim.x + threadIdx.x;
    if (t < n16) a[t] = fmaxf(a[t], 0.0f);
}

// ---------------- layer-2: z = hr @ W2 ; out = z*dis^2 + b2 then edge scatter ----------------

__global__ void k_lin2(const float* __restrict__ hr, const float* __restrict__ dis,
                       const float* __restrict__ W2, const float* __restrict__ b2,
                       float* __restrict__ z, float* __restrict__ out, int n) {
    int i = blockIdx.x * blockDim.x + threadIdx.x;
    if (i < n) {
        float acc = 0.0f;
#pragma unroll
        for (int j = 0; j < HID; ++j) acc += hr[i * HID + j] * W2[j];
        z[i] = acc;
        float d = dis[i];
        out[i] = acc * d * d + b2[0];        // self loop + bias; full overwrite of d_out
    }
}

__global__ void k_agg2(const int* __restrict__ src, const int* __restrict__ dst,
                       const float* __restrict__ norm, const float* __restrict__ z,
                       float* __restrict__ out, int e) {
    int i = blockIdx.x * blockDim.x + threadIdx.x;
    if (i < e) atomicAdd(&out[dst[i]], z[src[i]] * norm[i]);
}

// ---------------- launch ----------------

extern "C" void kernel_launch(void* const* d_in, const int* in_sizes, int n_in,
                              void* d_out, int out_size, void* d_ws, size_t ws_size,
                              hipStream_t stream) {
    const float* x  = (const float*)d_in[0];
    const int*   ei = (const int*)  d_in[1];
    const float* w  = (const float*)d_in[2];
    const float* W1 = (const float*)d_in[3];
    const float* b1 = (const float*)d_in[4];
    const float* W2 = (const float*)d_in[5];
    const float* b2 = (const float*)d_in[6];
    float* out = (float*)d_out;

    const int n = in_sizes[0] / IN_DIM;      // 100000
    const int e = in_sizes[2];               // 3200000
    const int* src = ei;
    const int* dst = ei + e;

    float* deg  = (float*)d_ws;              // n       (becomes dis in place)
    float* norm = deg + n;                   // e
    float* h    = norm + e;                  // n*16
    float* agg  = h + (size_t)n * HID;       // n*16
    float* z    = agg + (size_t)n * HID;     // n

    const int B = 256;
    k_init_deg<<<(n + B - 1) / B, B, 0, stream>>>(deg, n);
    k_deg     <<<(e + B - 1) / B, B, 0, stream>>>(dst, w, deg, e);
    k_rsqrt   <<<(n + B - 1) / B, B, 0, stream>>>(deg, n);
    k_norm    <<<(e + B - 1) / B, B, 0, stream>>>(src, dst, w, deg, norm, e);

    int n_tiles = n / 16;                    // 100000 % 16 == 0
    k_gemm1   <<<(n_tiles + 7) / 8, 256, 0, stream>>>(x, W1, h, n_tiles);

    int n16 = n * HID;
    k_agg1_init<<<(n16 + B - 1) / B, B, 0, stream>>>(h, deg, b1, agg, n16);
    {
        long tot = (long)e * HID;
        k_agg1 <<<(int)((tot + B - 1) / B), B, 0, stream>>>(src, dst, norm, h, agg, e);
    }
    k_relu    <<<(n16 + B - 1) / B, B, 0, stream>>>(agg, n16);
    k_lin2    <<<(n + B - 1) / B, B, 0, stream>>>(agg, deg, W2, b2, z, out, n);
    k_agg2    <<<(e + B - 1) / B, B, 0, stream>>>(src, dst, norm, z, out, e);
}